// MoPeAttention_54846732370521
// MI455X (gfx1250) — compile-verified
//
#include <hip/hip_runtime.h>
#include <hip/hip_bf16.h>
#include <math.h>

typedef __attribute__((ext_vector_type(2))) float v2f;
typedef __attribute__((ext_vector_type(8))) float v8f;

#define B_    4
#define T_    1024
#define D_    512
#define M_    64
#define BT_   4096
#define NOUT_ 320          // q(64) k(64) v(64) r_k(64) r_v(64)
#define K2_   1024         // 2*D

// ---- workspace layout (float offsets) ----
#define OFF_NORMX 0
#define SZ_NORMX  (B_*(T_+1)*D_)            // 2,099,200
#define OFF_WCT   (OFF_NORMX + SZ_NORMX)
#define SZ_WCT    (K2_*NOUT_)               // 327,680
#define OFF_P     (OFF_WCT + SZ_WCT)
#define SZ_P      (BT_*NOUT_)               // 1,310,720
#define OFF_CKK   (OFF_P + SZ_P)
#define SZ_CMAT   (BT_*M_*M_)               // 16,777,216
#define OFF_CVK   (OFF_CKK + SZ_CMAT)
#define OFF_DIAG  (OFF_CVK + SZ_CMAT)
#define SZ_DIAG   (BT_*M_)
#define OFF_OUTX  (OFF_DIAG + SZ_DIAG)
#define SZ_OUTX   (BT_*M_)
#define OFF_ACC   (OFF_OUTX + SZ_OUTX)      // total ~37.7M floats (~151 MB)

// ---- output layout (float offsets) ----
#define OUT_Y    0
#define OUT_CKK  (BT_*D_)                   // 2,097,152
#define OUT_CVK  (OUT_CKK + B_*M_*M_)
#define OUT_CVV  (OUT_CVK + B_*M_*M_)
#define OUT_LOSS (OUT_CVV + B_*M_*M_)       // 2,146,304

// ---- CDNA5 async global->LDS path (guarded: falls back if builtin absent) --
#if defined(__has_builtin)
#  if __has_builtin(__builtin_amdgcn_global_load_async_to_lds_b32)
#    define HAS_ASYNC_LDS 1
#  endif
#endif
#ifndef HAS_ASYNC_LDS
#  define HAS_ASYNC_LDS 0
#endif

#if HAS_ASYNC_LDS
typedef __attribute__((address_space(1))) int* gptr_i32;
typedef __attribute__((address_space(3))) int* lptr_i32;
__device__ __forceinline__ void async_ld_b32(const float* g, float* l) {
  __builtin_amdgcn_global_load_async_to_lds_b32(
      (gptr_i32)(uintptr_t)g, (lptr_i32)(uintptr_t)l,
      /*offset=*/0, /*cpol=*/0);
}
template <int N> __device__ __forceinline__ void wait_async() {
#  if __has_builtin(__builtin_amdgcn_s_wait_asynccnt)
  __builtin_amdgcn_s_wait_asynccnt(N);
#  else
  asm volatile("s_wait_asynccnt %0" ::"n"(N) : "memory");
#  endif
}
#endif

__device__ __forceinline__ v8f wmma_f32(v2f a, v2f b, v8f c) {
  // V_WMMA_F32_16X16X4_F32: D = A(16x4) * B(4x16) + C, fp32 throughout
  return __builtin_amdgcn_wmma_f32_16x16x4_f32(
      false, a, false, b, (short)0, c, false, false);
}

// ---------------------------------------------------------------- stage 0
__global__ void zero_acc_kernel(float* ws) {
  ws[OFF_ACC] = 0.0f;
}

__global__ __launch_bounds__(256) void ln_kernel(
    const float* __restrict__ x, const float* __restrict__ last_x,
    const float* __restrict__ ln_w, const float* __restrict__ ln_b,
    float* __restrict__ normx) {
  __shared__ float r1[256], r2[256];
  const int tid = threadIdx.x;
  const int rr  = blockIdx.x;              // 0 .. B*(T+1)-1
  const int b   = rr / (T_ + 1);
  const int s   = rr % (T_ + 1);
  const float* src = (s == 0) ? (last_x + (size_t)b * D_)
                              : (x + ((size_t)b * T_ + (s - 1)) * D_);
  float s1 = 0.f, s2 = 0.f;
  for (int d = tid; d < D_; d += 256) { float v = src[d]; s1 += v; s2 += v * v; }
  r1[tid] = s1; r2[tid] = s2; __syncthreads();
  for (int w = 128; w > 0; w >>= 1) {
    if (tid < w) { r1[tid] += r1[tid + w]; r2[tid] += r2[tid + w]; }
    __syncthreads();
  }
  const float mu   = r1[0] * (1.0f / D_);
  const float var  = r2[0] * (1.0f / D_) - mu * mu;
  const float rstd = rsqrtf(var + 1e-5f);
  float* dst = normx + (size_t)b * ((T_ + 1) * D_) + (size_t)s * D_;
  for (int d = tid; d < D_; d += 256) {
    dst[d] = (src[d] - mu) * rstd * ln_w[d] + ln_b[d];
  }
}

// Fold the per-projection lerp into the weights:
// WcT[k][o] = w[o, d]*(1-lw)  for k<D  (d=k),  w[o, d]*lw for k>=D (d=k-D)
__global__ __launch_bounds__(256) void wprep_kernel(
    const float* __restrict__ qkv_lerp_w, const float* __restrict__ qkv_w,
    const float* __restrict__ r_lerp_w,   const float* __restrict__ r_w,
    float* __restrict__ wct) {
  const int idx = blockIdx.x * 256 + threadIdx.x;
  if (idx >= K2_ * NOUT_) return;
  const int kk = idx / NOUT_;
  const int o  = idx % NOUT_;
  const bool hi = (kk >= D_);
  const int d = hi ? (kk - D_) : kk;
  float w, lwraw;
  if (o < 192) {                            // q, k, v
    const int p = o >> 6;
    const int row = o & 63;
    w     = qkv_w[((size_t)p * M_ + row) * D_ + d];
    lwraw = qkv_lerp_w[p * D_ + d];
  } else {                                   // r (2M outputs)
    w     = r_w[(size_t)(o - 192) * D_ + d];
    lwraw = r_lerp_w[d];
  }
  const float lw = 1.0f / (1.0f + expf(-lwraw));
  wct[idx] = w * (hi ? lw : (1.0f - lw));
}

// ---------------------------------------------------------------- stage 1
// P(4096 x 320) = A2(4096 x 1024) @ WcT ; A2 row bt is 1024 contiguous
// floats of normx starting at b*(T+1)*D + t*D. r-columns (>=192) get the
// affine transform 0.1 + 0.89*r in the epilogue.
__global__ __launch_bounds__(256) void gemm_qkvr_kernel(
    const float* __restrict__ normx, const float* __restrict__ wct,
    float* __restrict__ P) {
  __shared__ float As[64][33];
  __shared__ float Bs[32][65];
  const int tid = threadIdx.x;
  const int r0  = blockIdx.x * 64;          // bt tile (stays inside a batch)
  const int c0  = blockIdx.y * 64;          // output-col tile
  const int b   = r0 >> 10;
  const int t0  = r0 & 1023;
  const float* Abase = normx + (size_t)b * ((T_ + 1) * D_) + (size_t)t0 * D_;
  const int wv = tid >> 5, lane = tid & 31;
  const int m = lane & 15, h = lane >> 4;   // ISA A/B lane layout
  const int sr = wv & 3;                    // subtile row (x16)
  const int scb = (wv >> 2) * 2;            // two adjacent col subtiles
  v8f acc0 = {0,0,0,0,0,0,0,0}, acc1 = {0,0,0,0,0,0,0,0};

  for (int k0 = 0; k0 < K2_; k0 += 32) {
    for (int idx = tid; idx < 64 * 32; idx += 256) {
      const int r = idx >> 5, kk = idx & 31;
      As[r][kk] = Abase[(size_t)r * D_ + k0 + kk];
    }
    for (int idx = tid; idx < 32 * 64; idx += 256) {
      const int kk = idx >> 6, n = idx & 63;
      Bs[kk][n] = wct[(size_t)(k0 + kk) * NOUT_ + c0 + n];
    }
    __syncthreads();
#pragma unroll
    for (int kk = 0; kk < 32; kk += 4) {
      v2f a, b0v, b1v;
      a.x   = As[sr * 16 + m][kk + 2 * h];
      a.y   = As[sr * 16 + m][kk + 2 * h + 1];
      b0v.x = Bs[kk + 2 * h][scb * 16 + m];
      b0v.y = Bs[kk + 2 * h + 1][scb * 16 + m];
      b1v.x = Bs[kk + 2 * h][(scb + 1) * 16 + m];
      b1v.y = Bs[kk + 2 * h + 1][(scb + 1) * 16 + m];
      acc0 = wmma_f32(a, b0v, acc0);
      acc1 = wmma_f32(a, b1v, acc1);
    }
    __syncthreads();
  }
#pragma unroll
  for (int r = 0; r < 8; ++r) {             // D layout: M = r + 8*h
    const int grow = r0 + sr * 16 + r + 8 * h;
    const int ca = c0 + scb * 16 + m;
    const int cb = c0 + (scb + 1) * 16 + m;
    float va = acc0[r], vb = acc1[r];
    if (ca >= 192) va = 0.1f + 0.89f * va;  // R_MIN + (R_MAX-R_MIN)*r
    if (cb >= 192) vb = 0.1f + 0.89f * vb;
    P[(size_t)grow * NOUT_ + ca] = va;
    P[(size_t)grow * NOUT_ + cb] = vb;
  }
}

// ---------------------------------------------------------------- stage 2
// Elementwise scan c <- c + (d-c)*r over T, parallel over B*3*64*64 = 49152
// elements. k/v/r_k/r_v vectors staged through LDS 8 timesteps at a time,
// double-buffered with GLOBAL_LOAD_ASYNC_TO_LDS (ASYNCcnt) when available.
__global__ __launch_bounds__(256) void scan_kernel(
    const float* __restrict__ P,
    const float* __restrict__ lastCkk, const float* __restrict__ lastCvk,
    const float* __restrict__ lastCvv,
    float* __restrict__ ckk_ws, float* __restrict__ cvk_ws,
    float* __restrict__ diag_ws, float* __restrict__ out) {
  __shared__ float sm[2][8][256];           // cols: k[0:64) v[64:128) rk[128:192) rv[192:256)
  const int tid = threadIdx.x;
  const int b   = blockIdx.x / 48;
  const int g   = blockIdx.x % 48;
  const int e   = g * 256 + tid;            // element within batch
  const int mat = e >> 12;                  // 0=kk 1=vk 2=vv (uniform per block)
  const int ij  = e & 4095;
  const int i   = ij >> 6, j = ij & 63;

  int ai, bi, rai, rbi;
  if (mat == 0)      { ai = i;      bi = j;      rai = 128 + i; rbi = 128 + j; }
  else if (mat == 1) { ai = 64 + i; bi = j;      rai = 192 + i; rbi = 128 + j; }
  else               { ai = 64 + i; bi = 64 + j; rai = 192 + i; rbi = 192 + j; }

  float c = (mat == 0 ? lastCkk : (mat == 1 ? lastCvk : lastCvv))
                [(size_t)b * 4096 + ij];
  const float* Pb = P + (size_t)b * T_ * NOUT_;

#if HAS_ASYNC_LDS
  // prologue: chunk 0 into buffer 0
#pragma unroll
  for (int tl = 0; tl < 8; ++tl)
    async_ld_b32(Pb + (size_t)tl * NOUT_ + 64 + tid, &sm[0][tl][tid]);
  int pb = 0;
  for (int tc = 0; tc < T_; tc += 8, pb ^= 1) {
    if (tc + 8 < T_) {
#pragma unroll
      for (int tl = 0; tl < 8; ++tl)
        async_ld_b32(Pb + (size_t)(tc + 8 + tl) * NOUT_ + 64 + tid,
                     &sm[pb ^ 1][tl][tid]);
      wait_async<8>();                      // chunk tc done; next in flight
    } else {
      wait_async<0>();
    }
    __syncthreads();
#pragma unroll
    for (int tl = 0; tl < 8; ++tl) {
      const float d = sm[pb][tl][ai] * sm[pb][tl][bi];
      const float r = sm[pb][tl][rai] * sm[pb][tl][rbi];
      c = fmaf(r, d - c, c);
      const size_t o = ((size_t)b * T_ + (tc + tl)) * 4096 + ij;
      if (mat == 0)      ckk_ws[o] = c;
      else if (mat == 1) cvk_ws[o] = c;
      else if (i == j)   diag_ws[((size_t)b * T_ + (tc + tl)) * 64 + i] = c;
    }
    __syncthreads();                        // done reading buf pb before reuse
  }
#else
  for (int tc = 0; tc < T_; tc += 8) {
    __syncthreads();
#pragma unroll
    for (int tl = 0; tl < 8; ++tl)
      sm[0][tl][tid] = Pb[(size_t)(tc + tl) * NOUT_ + 64 + tid];
    if (tc + 8 < T_)
      __builtin_prefetch(Pb + (size_t)(tc + 8) * NOUT_ + 64 + tid, 0, 1);
    __syncthreads();
#pragma unroll
    for (int tl = 0; tl < 8; ++tl) {
      const float d = sm[0][tl][ai] * sm[0][tl][bi];
      const float r = sm[0][tl][rai] * sm[0][tl][rbi];
      c = fmaf(r, d - c, c);
      const size_t o = ((size_t)b * T_ + (tc + tl)) * 4096 + ij;
      if (mat == 0)      ckk_ws[o] = c;
      else if (mat == 1) cvk_ws[o] = c;
      else if (i == j)   diag_ws[((size_t)b * T_ + (tc + tl)) * 64 + i] = c;
    }
  }
#endif
  // final states to d_out
  if (mat == 0)      out[OUT_CKK + (size_t)b * 4096 + ij] = c;
  else if (mat == 1) out[OUT_CVK + (size_t)b * 4096 + ij] = c;
  else               out[OUT_CVV + (size_t)b * 4096 + ij] = c;
}

// ---------------------------------------------------------------- stage 3
// Per (b,t): solve C_kk * X = C_vk^T (C_kk symmetric) by Gauss-Jordan in
// LDS; X = memory^T. Then out_x = memory@q and the resid-trace partial.
#define AS_ 132                              // padded row stride of Aug
__global__ __launch_bounds__(256) void solve_kernel(
    const float* __restrict__ ckk_ws, const float* __restrict__ cvk_ws,
    const float* __restrict__ diag_ws, const float* __restrict__ P,
    float* __restrict__ outx_ws, float* __restrict__ accp) {
  __shared__ float Aug[64 * AS_];
  __shared__ float Cvkc[64 * 64];
  __shared__ float qv[64], dvv[64], fcol[64], red[256];
  const int tid = threadIdx.x;
  const size_t bt = blockIdx.x;
  const size_t cb = bt * 4096;

  // load: left block = C_kk, right block = C_vk^T (contiguous global reads)
  for (int idx = tid; idx < 4096; idx += 256) {
    const int l = idx >> 6, cc = idx & 63;
    Aug[l * AS_ + cc] = ckk_ws[cb + idx];
  }
  for (int idx = tid; idx < 4096; idx += 256) {
    const int i = idx >> 6, l = idx & 63;
    const float v = cvk_ws[cb + idx];
    Cvkc[idx] = v;
    Aug[l * AS_ + 64 + i] = v;               // transpose into RHS
  }
  if (tid < 64) { qv[tid] = P[bt * NOUT_ + tid]; dvv[tid] = diag_ws[bt * 64 + tid]; }
  __syncthreads();

  for (int p = 0; p < 64; ++p) {
    const float piv = Aug[p * AS_ + p];
    if (tid < 64) fcol[tid] = (tid == p) ? 0.0f : Aug[tid * AS_ + p];
    __syncthreads();
    const float invp = 1.0f / piv;
    if (tid < 128) Aug[p * AS_ + tid] *= invp;
    __syncthreads();
#pragma unroll
    for (int s = 0; s < 32; ++s) {           // 8192 elements / 256 threads
      const int ee = tid + (s << 8);
      const int r = ee >> 7, cc = ee & 127;
      if (r != p)
        Aug[r * AS_ + cc] = fmaf(-fcol[r], Aug[p * AS_ + cc], Aug[r * AS_ + cc]);
    }
    __syncthreads();
  }

  // out_x[i] = sum_l memory[i][l]*q[l] = sum_l Aug[l][64+i]*q[l]
  if (tid < 64) {
    float s = 0.f;
    for (int l = 0; l < 64; ++l) s = fmaf(Aug[l * AS_ + 64 + tid], qv[l], s);
    outx_ws[bt * 64 + tid] = s;
  }
  // trace(resid) = sum_i diag(C_vv) - sum_{i,l} memory[i][l]*C_vk[i][l]
  float part = 0.f;
#pragma unroll
  for (int s = 0; s < 16; ++s) {
    const int ee = tid + (s << 8);
    const int i = ee >> 6, l = ee & 63;
    part = fmaf(Aug[l * AS_ + 64 + i], Cvkc[i * 64 + l], part);
  }
  red[tid] = part; __syncthreads();
  for (int w = 128; w > 0; w >>= 1) {
    if (tid < w) red[tid] += red[tid + w];
    __syncthreads();
  }
  if (tid == 0) {
    float ds = 0.f;
    for (int i = 0; i < 64; ++i) ds += dvv[i];
    atomicAdd(accp, ds - red[0]);
  }
}

// ---------------------------------------------------------------- stage 4
// y = x + out_x(4096x64) @ out_w^T(64x512) + out_b
__global__ __launch_bounds__(256) void gemm_out_kernel(
    const float* __restrict__ outx_ws, const float* __restrict__ out_w,
    const float* __restrict__ out_b, const float* __restrict__ x,
    float* __restrict__ out) {
  __shared__ float As[64][65];
  __shared__ float Bs[64][65];
  const int tid = threadIdx.x;
  const int r0 = blockIdx.x * 64;           // bt tile
  const int d0 = blockIdx.y * 64;           // d tile
  for (int idx = tid; idx < 4096; idx += 256) {
    const int r = idx >> 6, mm = idx & 63;
    As[r][mm] = outx_ws[(size_t)(r0 + r) * 64 + mm];
  }
  for (int idx = tid; idx < 4096; idx += 256) {
    const int dl = idx >> 6, mm = idx & 63;
    Bs[mm][dl] = out_w[(size_t)(d0 + dl) * 64 + mm];   // transpose in LDS
  }
  __syncthreads();
  const int wv = tid >> 5, lane = tid & 31;
  const int m = lane & 15, h = lane >> 4;
  const int sr = wv & 3;
  const int scb = (wv >> 2) * 2;
  v8f acc0 = {0,0,0,0,0,0,0,0}, acc1 = {0,0,0,0,0,0,0,0};
#pragma unroll
  for (int kk = 0; kk < 64; kk += 4) {
    v2f a, b0v, b1v;
    a.x   = As[sr * 16 + m][kk + 2 * h];
    a.y   = As[sr * 16 + m][kk + 2 * h + 1];
    b0v.x = Bs[kk + 2 * h][scb * 16 + m];
    b0v.y = Bs[kk + 2 * h + 1][scb * 16 + m];
    b1v.x = Bs[kk + 2 * h][(scb + 1) * 16 + m];
    b1v.y = Bs[kk + 2 * h + 1][(scb + 1) * 16 + m];
    acc0 = wmma_f32(a, b0v, acc0);
    acc1 = wmma_f32(a, b1v, acc1);
  }
#pragma unroll
  for (int r = 0; r < 8; ++r) {
    const int grow = r0 + sr * 16 + r + 8 * h;
    const int ca = d0 + scb * 16 + m;
    const int cb2 = d0 + (scb + 1) * 16 + m;
    out[OUT_Y + (size_t)grow * D_ + ca]  = x[(size_t)grow * D_ + ca]  + acc0[r] + out_b[ca];
    out[OUT_Y + (size_t)grow * D_ + cb2] = x[(size_t)grow * D_ + cb2] + acc1[r] + out_b[cb2];
  }
}

__global__ void finalize_kernel(const float* ws, float* out) {
  out[OUT_LOSS] = fabsf(ws[OFF_ACC] * (1.0f / (float)BT_));
}

// ----------------------------------------------------------------
extern "C" void kernel_launch(void* const* d_in, const int* in_sizes, int n_in,
                              void* d_out, int out_size, void* d_ws, size_t ws_size,
                              hipStream_t stream) {
  const float* x          = (const float*)d_in[0];
  const float* last_x     = (const float*)d_in[1];
  const float* lastCkk    = (const float*)d_in[2];
  const float* lastCvk    = (const float*)d_in[3];
  const float* lastCvv    = (const float*)d_in[4];
  const float* ln_w       = (const float*)d_in[5];
  const float* ln_b       = (const float*)d_in[6];
  const float* qkv_lerp_w = (const float*)d_in[7];
  const float* qkv_w      = (const float*)d_in[8];
  const float* r_lerp_w   = (const float*)d_in[9];
  const float* r_w        = (const float*)d_in[10];
  const float* out_w      = (const float*)d_in[11];
  const float* out_b      = (const float*)d_in[12];
  float* out = (float*)d_out;
  float* ws  = (float*)d_ws;

  float* normx   = ws + OFF_NORMX;
  float* wct     = ws + OFF_WCT;
  float* P       = ws + OFF_P;
  float* ckk_ws  = ws + OFF_CKK;
  float* cvk_ws  = ws + OFF_CVK;
  float* diag_ws = ws + OFF_DIAG;
  float* outx_ws = ws + OFF_OUTX;
  float* accp    = ws + OFF_ACC;

  zero_acc_kernel<<<1, 1, 0, stream>>>(ws);
  ln_kernel<<<B_ * (T_ + 1), 256, 0, stream>>>(x, last_x, ln_w, ln_b, normx);
  wprep_kernel<<<(K2_ * NOUT_) / 256, 256, 0, stream>>>(qkv_lerp_w, qkv_w,
                                                        r_lerp_w, r_w, wct);
  gemm_qkvr_kernel<<<dim3(BT_ / 64, NOUT_ / 64), 256, 0, stream>>>(normx, wct, P);
  scan_kernel<<<B_ * 48, 256, 0, stream>>>(P, lastCkk, lastCvk, lastCvv,
                                           ckk_ws, cvk_ws, diag_ws, out);
  solve_kernel<<<BT_, 256, 0, stream>>>(ckk_ws, cvk_ws, diag_ws, P, outx_ws, accp);
  gemm_out_kernel<<<dim3(BT_ / 64, D_ / 64), 256, 0, stream>>>(outx_ws, out_w,
                                                               out_b, x, out);
  finalize_kernel<<<1, 1, 0, stream>>>(ws, out);
}